// DCConvResNet_43568148251273
// MI455X (gfx1250) — compile-verified
//
#include <hip/hip_runtime.h>
#include <math.h>

typedef __attribute__((ext_vector_type(2))) float v2f;
typedef __attribute__((ext_vector_type(8))) float v8f;

// ---------------------------------------------------------------------------
// KNN: one 128-thread workgroup per query. Each thread keeps a sorted top-16
// over a strided slice of the sources, then pairwise merge in LDS.
// ---------------------------------------------------------------------------
#define KNN_THREADS 128

__global__ __launch_bounds__(KNN_THREADS) void knn_kernel(
    const float* __restrict__ qpos, const float* __restrict__ spos,
    int ns, int k, int* __restrict__ outIdx)
{
    __shared__ float sd[KNN_THREADS * 16];
    __shared__ int   si[KNN_THREADS * 16];

    const int q = blockIdx.x;
    const int tid = threadIdx.x;

    const float qx = qpos[q * 3 + 0];
    const float qy = qpos[q * 3 + 1];
    const float qz = qpos[q * 3 + 2];
    const float qq = qx * qx + qy * qy + qz * qz;

    float ld[16];
    int   li[16];
#pragma unroll
    for (int i = 0; i < 16; i++) { ld[i] = __builtin_inff(); li[i] = 0x7fffffff; }

    for (int j = tid; j < ns; j += KNN_THREADS) {
        const float sx = spos[j * 3 + 0];
        const float sy = spos[j * 3 + 1];
        const float sz = spos[j * 3 + 2];
        const float ss = sx * sx + sy * sy + sz * sz;
        const float d  = qq + ss - 2.0f * (qx * sx + qy * sy + qz * sz);
        // branchless sorted insert (ascending by (d, idx))
        float cd = d; int ci = j;
#pragma unroll
        for (int p = 0; p < 16; p++) {
            const bool lt = (cd < ld[p]) || (cd == ld[p] && ci < li[p]);
            const float td = lt ? ld[p] : cd;
            const int   ti = lt ? li[p] : ci;
            ld[p] = lt ? cd : ld[p];
            li[p] = lt ? ci : li[p];
            cd = td; ci = ti;
        }
    }

#pragma unroll
    for (int i = 0; i < 16; i++) { sd[tid * 16 + i] = ld[i]; si[tid * 16 + i] = li[i]; }
    __syncthreads();

    for (int s = KNN_THREADS / 2; s >= 1; s >>= 1) {
        float od[16]; int oi[16];
        const bool act = tid < s;
        if (act) {
            int ia = 0, ib = 0;
#pragma unroll
            for (int o = 0; o < 16; o++) {
                const float d1 = sd[tid * 16 + ia];
                const int   i1 = si[tid * 16 + ia];
                const float d2 = sd[(tid + s) * 16 + ib];
                const int   i2 = si[(tid + s) * 16 + ib];
                const bool ta = (ib >= 16) ||
                                ((ia < 16) && ((d1 < d2) || (d1 == d2 && i1 <= i2)));
                od[o] = ta ? d1 : d2;
                oi[o] = ta ? i1 : i2;
                ia += ta ? 1 : 0;
                ib += ta ? 0 : 1;
            }
        }
        __syncthreads();
        if (act) {
#pragma unroll
            for (int o = 0; o < 16; o++) { sd[tid * 16 + o] = od[o]; si[tid * 16 + o] = oi[o]; }
        }
        __syncthreads();
    }

    if (tid < k) outIdx[q * k + tid] = si[tid];
}

// ---------------------------------------------------------------------------
// DCConv stage 1: per query build normalized local frame (PCA via 3x3 Jacobi),
// polynomial basis phi (45), gather neighbor features, emit
// T[q, b*cin + c] = (1/k) * sum_j phi[j][b] * nf[j][c].
// ---------------------------------------------------------------------------
#define DC_THREADS 128

__global__ __launch_bounds__(DC_THREADS) void dcconv_t_kernel(
    const float* __restrict__ qpos, const float* __restrict__ spos,
    const float* __restrict__ sfeat, const int* __restrict__ nidx,
    int k, int cin, int Kdim, float* __restrict__ T)
{
    __shared__ int   idxS[16];
    __shared__ float rS[16][3];
    __shared__ float n2S[16];
    __shared__ float phiS[16][45];
    __shared__ float nfS[16 * 256];
    __shared__ float vecsS[9];
    __shared__ float invScaleS;

    const int q = blockIdx.x;
    const int tid = threadIdx.x;

    if (tid < k) idxS[tid] = nidx[q * k + tid];
    __syncthreads();

    const float qx = qpos[q * 3 + 0];
    const float qy = qpos[q * 3 + 1];
    const float qz = qpos[q * 3 + 2];

    if (tid < k) {
        const int j = idxS[tid];
        const float rx = spos[j * 3 + 0] - qx;
        const float ry = spos[j * 3 + 1] - qy;
        const float rz = spos[j * 3 + 2] - qz;
        rS[tid][0] = rx; rS[tid][1] = ry; rS[tid][2] = rz;
        n2S[tid] = rx * rx + ry * ry + rz * rz;
    }
    __syncthreads();

    if (tid == 0) {
        float mx = 0.0f;
        for (int j = 0; j < k; j++) mx = fmaxf(mx, sqrtf(n2S[j]));
        invScaleS = 1.0f / (mx + 1e-8f);
    }
    __syncthreads();

    if (tid < k) {
        const float iv = invScaleS;
        rS[tid][0] *= iv; rS[tid][1] *= iv; rS[tid][2] *= iv;
    }
    __syncthreads();

    if (tid == 0) {
        // covariance
        float c00 = 0, c01 = 0, c02 = 0, c11 = 0, c12 = 0, c22 = 0;
        for (int j = 0; j < k; j++) {
            const float x = rS[j][0], y = rS[j][1], z = rS[j][2];
            c00 += x * x; c01 += x * y; c02 += x * z;
            c11 += y * y; c12 += y * z; c22 += z * z;
        }
        const float ik = 1.0f / (float)k;
        float A[3][3] = { {c00 * ik, c01 * ik, c02 * ik},
                          {c01 * ik, c11 * ik, c12 * ik},
                          {c02 * ik, c12 * ik, c22 * ik} };
        float V[3][3] = { {1, 0, 0}, {0, 1, 0}, {0, 0, 1} };
        // cyclic Jacobi sweeps
        for (int sw = 0; sw < 12; sw++) {
            for (int pp = 0; pp < 3; pp++) {
                const int p  = (pp == 2) ? 1 : 0;
                const int qi = (pp == 0) ? 1 : 2;
                const float apq = A[p][qi];
                if (fabsf(apq) <= 1e-18f) continue;
                const float theta = (A[qi][qi] - A[p][p]) / (2.0f * apq);
                const float t = (theta >= 0.0f ? 1.0f : -1.0f) /
                                (fabsf(theta) + sqrtf(theta * theta + 1.0f));
                const float cc = 1.0f / sqrtf(t * t + 1.0f);
                const float s  = t * cc;
                for (int i = 0; i < 3; i++) {
                    const float aip = A[i][p], aiq = A[i][qi];
                    A[i][p]  = cc * aip - s * aiq;
                    A[i][qi] = s  * aip + cc * aiq;
                }
                for (int i = 0; i < 3; i++) {
                    const float api = A[p][i], aqi = A[qi][i];
                    A[p][i]  = cc * api - s * aqi;
                    A[qi][i] = s  * api + cc * aqi;
                }
                for (int i = 0; i < 3; i++) {
                    const float vip = V[i][p], viq = V[i][qi];
                    V[i][p]  = cc * vip - s * viq;
                    V[i][qi] = s  * vip + cc * viq;
                }
            }
        }
        float ev[3] = { A[0][0], A[1][1], A[2][2] };
        int ord[3] = { 0, 1, 2 };
        if (ev[ord[0]] > ev[ord[1]]) { int t0 = ord[0]; ord[0] = ord[1]; ord[1] = t0; }
        if (ev[ord[1]] > ev[ord[2]]) { int t0 = ord[1]; ord[1] = ord[2]; ord[2] = t0; }
        if (ev[ord[0]] > ev[ord[1]]) { int t0 = ord[0]; ord[0] = ord[1]; ord[1] = t0; }
        for (int jj = 0; jj < 3; jj++) {
            const int cI = ord[jj];
            float v0 = V[0][cI], v1 = V[1][cI], v2 = V[2][cI];
            float m = v0;
            if (fabsf(v1) > fabsf(m)) m = v1;
            if (fabsf(v2) > fabsf(m)) m = v2;
            const float sg = (m < 0.0f) ? -1.0f : 1.0f;
            vecsS[0 * 3 + jj] = v0 * sg;
            vecsS[1 * 3 + jj] = v1 * sg;
            vecsS[2 * 3 + jj] = v2 * sg;
        }
    }
    __syncthreads();

    if (tid < k) {
        const float x = rS[tid][0], y = rS[tid][1], z = rS[tid][2];
        const float xr = x * vecsS[0] + y * vecsS[3] + z * vecsS[6];
        const float yr = x * vecsS[1] + y * vecsS[4] + z * vecsS[7];
        const float zr = x * vecsS[2] + y * vecsS[5] + z * vecsS[8];
        float px[5], py[3], pz[3];
        px[0] = 1.0f;
#pragma unroll
        for (int i = 1; i < 5; i++) px[i] = px[i - 1] * xr;
        py[0] = 1.0f; py[1] = yr; py[2] = yr * yr;
        pz[0] = 1.0f; pz[1] = zr; pz[2] = zr * zr;
#pragma unroll
        for (int ix = 0; ix < 5; ix++)
#pragma unroll
            for (int iy = 0; iy < 3; iy++)
#pragma unroll
                for (int iz = 0; iz < 3; iz++)
                    phiS[tid][(ix * 3 + iy) * 3 + iz] = px[ix] * py[iy] * pz[iz];
    }

    // gather neighbor features
    for (int t = tid; t < k * cin; t += DC_THREADS) {
        const int j = t / cin;
        const int c = t - j * cin;
        nfS[j * cin + c] = sfeat[(size_t)idxS[j] * cin + c];
    }
    __syncthreads();

    const float invk = 1.0f / (float)k;
    for (int t = tid; t < Kdim; t += DC_THREADS) {
        const int b = t / cin;
        const int c = t - b * cin;
        float s = 0.0f;
        for (int j = 0; j < k; j++) s += phiS[j][b] * nfS[j * cin + c];
        T[(size_t)q * Kdim + t] = s * invk;
    }
}

// ---------------------------------------------------------------------------
// WMMA fp32 GEMM: C(MxN) = relu( A(MxK) @ B(KxN) [+ bias] [+ addend] )
// One wave32 per 16x16 output tile, v_wmma_f32_16x16x4_f32, K unrolled by 8
// with two independent accumulators. Out-of-range A rows are clamped (their
// accumulator rows are garbage but never stored). K % 4 == 0, N % 16 == 0.
// ---------------------------------------------------------------------------
__global__ __launch_bounds__(256) void gemm_wmma_kernel(
    const float* __restrict__ A, int lda,
    const float* __restrict__ B, int N,
    const float* __restrict__ bias, const float* __restrict__ addend,
    float* __restrict__ C, int M, int K, int tilesN, int totalTiles)
{
    const int w = blockIdx.x * 8 + (threadIdx.x >> 5);
    if (w >= totalTiles) return;   // wave-uniform exit; EXEC stays full for WMMA

    const int tm = w / tilesN;
    const int tn = w - tm * tilesN;
    const int lane = threadIdx.x & 31;
    const int l    = lane & 15;
    const int half = lane >> 4;

    const int r = tm * 16 + l;                 // logical A row for this lane
    const int rc = (r < M) ? r : (M - 1);      // clamped: always in-bounds
    const int c = tn * 16 + l;                 // B/C column for this lane

    // A: this lane supplies K=(kk+2*half, kk+2*half+1) -> one aligned b64 load.
    const float* pa = A + (size_t)rc * lda + 2 * half;
    // B: this lane supplies column c of rows (kk+2*half, kk+2*half+1).
    const float* pb = B + (size_t)(2 * half) * N + c;
    const size_t sN = (size_t)N;

    v8f acc0 = {0.f, 0.f, 0.f, 0.f, 0.f, 0.f, 0.f, 0.f};
    v8f acc1 = {0.f, 0.f, 0.f, 0.f, 0.f, 0.f, 0.f, 0.f};

    int kk = 0;
    for (; kk + 8 <= K; kk += 8) {
        const v2f a0 = *(const v2f*)(pa);
        const v2f a1 = *(const v2f*)(pa + 4);
        v2f b0, b1;
        b0.x = pb[0];
        b0.y = pb[sN];
        b1.x = pb[4 * sN];
        b1.y = pb[5 * sN];
        pa += 8;
        pb += 8 * sN;
        acc0 = __builtin_amdgcn_wmma_f32_16x16x4_f32(
            false, a0, false, b0, (short)0, acc0, false, false);
        acc1 = __builtin_amdgcn_wmma_f32_16x16x4_f32(
            false, a1, false, b1, (short)0, acc1, false, false);
    }
    if (kk < K) {  // one remaining chunk of 4 (K % 8 == 4)
        const v2f a0 = *(const v2f*)(pa);
        v2f b0;
        b0.x = pb[0];
        b0.y = pb[sN];
        acc0 = __builtin_amdgcn_wmma_f32_16x16x4_f32(
            false, a0, false, b0, (short)0, acc0, false, false);
    }

    const float bv = bias ? bias[c] : 0.0f;
#pragma unroll
    for (int v = 0; v < 8; v++) {
        const int row = tm * 16 + v + 8 * half;
        if (row < M) {
            float y = acc0[v] + acc1[v] + bv;
            if (addend) y += addend[(size_t)row * N + c];
            y = fmaxf(y, 0.0f);
            C[(size_t)row * N + c] = y;
        }
    }
}

// ---------------------------------------------------------------------------
// Head: out[0] = dot(feat[512], head_w[512]) + head_b
// ---------------------------------------------------------------------------
__global__ __launch_bounds__(256) void head_kernel(
    const float* __restrict__ f, const float* __restrict__ hw,
    const float* __restrict__ hb, float* __restrict__ out)
{
    __shared__ float red[256];
    const int tid = threadIdx.x;
    float s = 0.0f;
    for (int i = tid; i < 512; i += 256) s += f[i] * hw[i];
    red[tid] = s;
    __syncthreads();
    for (int st = 128; st > 0; st >>= 1) {
        if (tid < st) red[tid] += red[tid + st];
        __syncthreads();
    }
    if (tid == 0) out[0] = red[0] + hb[0];
}

// ---------------------------------------------------------------------------
// Host orchestration
// ---------------------------------------------------------------------------
extern "C" void kernel_launch(void* const* d_in, const int* in_sizes, int n_in,
                              void* d_out, int out_size, void* d_ws, size_t ws_size,
                              hipStream_t stream)
{
    (void)in_sizes; (void)n_in; (void)out_size; (void)ws_size;

    const float* pos  = (const float*)d_in[0];   // (32768,3)
    const float* chan = (const float*)d_in[1];   // (32768,4)
    const float* W[8];
    const float* Bv[8];
    for (int li = 1; li <= 7; li++) {
        W[li]  = (const float*)d_in[2 * li];
        Bv[li] = (const float*)d_in[2 * li + 1];
    }
    const float* p1 = (const float*)d_in[16];
    const float* p2 = (const float*)d_in[17];
    const float* p3 = (const float*)d_in[18];
    const float* p4 = (const float*)d_in[19];
    const float* hw = (const float*)d_in[20];
    const float* hb = (const float*)d_in[21];
    float* out = (float*)d_out;

    // reduction schedule (round(32768 * r^(i+1)), r = 2^(-15/4))
    const int N1 = 2435, N2 = 181, N3 = 13;  // N4 = 1

    // workspace layout
    char* wsb = (char*)d_ws;
    float* T   = (float*)wsb;                                  // 2435*2880 f32 = 28,051,200 B
    int*  idxb = (int*)(wsb + 28051200);                       // 2435*16 ints
    float* buf0 = (float*)(wsb + 28051200 + 160000);           // 155,840 f32 each
    float* buf1 = buf0 + 155840;
    float* buf2 = buf1 + 155840;

    auto knn = [&](int nq, int ns, int k) {
        knn_kernel<<<nq, KNN_THREADS, 0, stream>>>(pos, pos, ns, k, idxb);
    };
    auto dconv = [&](int nq, int k, int cin, const float* sfeat) {
        dcconv_t_kernel<<<nq, DC_THREADS, 0, stream>>>(pos, pos, sfeat, idxb,
                                                       k, cin, 45 * cin, T);
    };
    auto gemm = [&](const float* A, int lda, const float* Bm, int N,
                    const float* bias, const float* addend, float* C,
                    int M, int K) {
        const int tilesM = (M + 15) / 16;
        const int tilesN = N / 16;
        const int tiles  = tilesM * tilesN;
        const int blocks = (tiles + 7) / 8;
        gemm_wmma_kernel<<<blocks, 256, 0, stream>>>(A, lda, Bm, N, bias, addend,
                                                     C, M, K, tilesN, tiles);
    };

    // ---- Block 0: cfg [4,64,64], n_sel = 2435 ----
    knn(N1, 32768, 16);
    dconv(N1, 16, 4, chan);
    gemm(T, 180, W[1], 64, Bv[1], nullptr, buf0 /*x1*/, N1, 180);
    knn(N1, N1, 16);
    dconv(N1, 16, 64, buf0);
    gemm(T, 2880, W[2], 64, Bv[2], nullptr, buf1 /*x2*/, N1, 2880);
    gemm(chan, 4, p1, 64, nullptr, buf1, buf0 /*featB0 = relu(x2 + chan@p1)*/, N1, 4);

    // ---- Block 1: cfg [64,128,128], n_sel = 181 ----
    knn(N2, N1, 16);
    dconv(N2, 16, 64, buf0);
    gemm(T, 2880, W[3], 128, Bv[3], nullptr, buf1 /*x3*/, N2, 2880);
    knn(N2, N2, 16);
    dconv(N2, 16, 128, buf1);
    gemm(T, 5760, W[4], 128, Bv[4], nullptr, buf2 /*x4*/, N2, 5760);
    gemm(buf0, 64, p2, 128, nullptr, buf2, buf1 /*featB1*/, N2, 64);

    // ---- Block 2: cfg [128,256,256], n_sel = 13 ----
    knn(N3, N2, 16);
    dconv(N3, 16, 128, buf1);
    gemm(T, 5760, W[5], 256, Bv[5], nullptr, buf2 /*x5*/, N3, 5760);
    knn(N3, N3, 13);
    dconv(N3, 13, 256, buf2);
    gemm(T, 11520, W[6], 256, Bv[6], nullptr, buf0 /*x6*/, N3, 11520);
    gemm(buf1, 128, p3, 256, nullptr, buf0, buf2 /*featB2*/, N3, 128);

    // ---- Block 3: cfg [256,512], n_sel = 1 ----
    knn(1, N3, 13);
    dconv(1, 13, 256, buf2);
    gemm(T, 11520, W[7], 512, Bv[7], nullptr, buf1 /*x7*/, 1, 11520);
    gemm(buf2, 256, p4, 512, nullptr, buf1, buf0 /*featB3*/, 1, 256);

    // ---- Head ----
    head_kernel<<<1, 256, 0, stream>>>(buf0, hw, hb, out);
}